// MultiHeadedAttention_42640435315389
// MI455X (gfx1250) — compile-verified
//
#include <hip/hip_runtime.h>
#include <math.h>

typedef __attribute__((ext_vector_type(2))) float v2f;
typedef __attribute__((ext_vector_type(8))) float v8f;

// workspace layout (float offsets)
#define WS_Q 0u
#define WS_K 8388608u
#define WS_V 16777216u
#define WS_A 25165824u          // attention output, [16,512,32,32]
#define HEAD_STRIDE 2097152u    // b*N*D = 4*524288 floats per head

__device__ __forceinline__ v8f wmma_f32(v2f a, v2f b, v8f c) {
  // V_WMMA_F32_16X16X4_F32: (neg_a, A, neg_b, B, c_mod, C, reuse_a, reuse_b)
  return __builtin_amdgcn_wmma_f32_16x16x4_f32(false, a, false, b, (short)0, c,
                                               false, false);
}

// ---------------------------------------------------------------------------
// Kernel 1: 1x1-conv projection [512x128]x[128x1024] per image, fused window
// scatter into per-head windowed layout Qw[b, N_i, D_i].
// Each wave computes a 32(o) x 32(pix) block as 2x2 WMMA tiles.
// grid = (32 pixel blocks, 16 o blocks, 16 images), block = 32 (1 wave).
// ---------------------------------------------------------------------------
__global__ __launch_bounds__(32)
void qkv_proj_kernel(const float* __restrict__ x, const float* __restrict__ W,
                     const float* __restrict__ bias, float* __restrict__ dst) {
  const int lane = threadIdx.x & 31;
  const int m = lane & 15, hf = lane >> 4;
  const int p0 = blockIdx.x * 32;
  const int o0 = blockIdx.y * 32;
  const int n  = blockIdx.z;

  const float* xr    = x + (size_t)n * 128 * 1024;
  const float* arow0 = W + (size_t)(o0 + m) * 128 + 2 * hf;
  const float* arow1 = arow0 + 16 * 128;

  v8f acc00 = {}, acc01 = {}, acc10 = {}, acc11 = {};
  #pragma unroll 2
  for (int k = 0; k < 128; k += 4) {
    v2f a0 = *(const v2f*)(arow0 + k);
    v2f a1 = *(const v2f*)(arow1 + k);
    const float* bp = xr + (size_t)(k + 2 * hf) * 1024 + p0 + m;
    v2f b0, b1;
    b0.x = bp[0];
    b0.y = bp[1024];
    b1.x = bp[16];
    b1.y = bp[1024 + 16];
    acc00 = wmma_f32(a0, b0, acc00);
    acc01 = wmma_f32(a0, b1, acc01);
    acc10 = wmma_f32(a1, b0, acc10);
    acc11 = wmma_f32(a1, b1, acc11);
  }

  const int bb = n >> 2, tt = n & 3;
  #pragma unroll
  for (int pt = 0; pt < 2; ++pt) {
    const int p = p0 + 16 * pt + m;
    const int y = p >> 5, xw = p & 31;
    #pragma unroll
    for (int ot = 0; ot < 2; ++ot) {
      const v8f& acc = ot ? (pt ? acc11 : acc10) : (pt ? acc01 : acc00);
      #pragma unroll
      for (int v = 0; v < 8; ++v) {
        const int o = o0 + 16 * ot + v + 8 * hf;
        const int head = o >> 7, c = o & 127;
        const int logH = 3 - head;               // patch = 8>>head (square)
        const int s = 5 - logH;                  // log2(oh)
        const int oh_y = y >> logH, hy = y & ((1 << logH) - 1);
        const int ow_x = xw >> logH, wx = xw & ((1 << logH) - 1);
        const int row = (((tt << s) + oh_y) << s) + ow_x;
        const int col = (c << (2 * logH)) + (hy << logH) + wx;
        const int N = 4 << (2 * s);
        const int D = 128 << (2 * logH);
        const size_t idx =
            (size_t)head * HEAD_STRIDE + ((size_t)bb * N + row) * D + col;
        dst[idx] = acc[v] + bias[o];
      }
    }
  }
}

// ---------------------------------------------------------------------------
// Kernel 2: windowed attention for one head.
// grid = (N/16 query tiles, 4 batches), block = 128 (4 waves).
// LDS: S[16][N] scores + reduction scratch.
// ---------------------------------------------------------------------------
__global__ __launch_bounds__(128)
void attn_kernel(const float* __restrict__ Q, const float* __restrict__ K,
                 const float* __restrict__ V, float* __restrict__ out,
                 int N, int D, int head, int logH, float scale) {
  extern __shared__ float smem[];
  float* S    = smem;             // 16*N
  float* red  = smem + 16 * N;    // 128
  float* rmax = red + 128;        // 16
  float* rinv = rmax + 16;        // 16

  const int tid  = threadIdx.x;
  const int wave = tid >> 5;
  const int lane = tid & 31;
  const int m = lane & 15, hf = lane >> 4;
  const int q0 = blockIdx.x * 16;
  const int bb = blockIdx.y;

  const float* Qb = Q + (size_t)bb * N * D;
  const float* Kb = K + (size_t)bb * N * D;
  const float* Vb = V + (size_t)bb * N * D;

  // ---- phase 1: S = scale * Q K^T (paired column tiles) --------------
  const float* qrow = Qb + (size_t)(q0 + m) * D + 2 * hf;
  const int coltiles = N >> 4;   // always even (4/16/64/256)
  for (int jt = wave * 2; jt < coltiles; jt += 8) {
    const int j0 = jt << 4;
    const float* krow0 = Kb + (size_t)(j0 + m) * D + 2 * hf;
    const float* krow1 = krow0 + (size_t)16 * D;
    v8f acc0 = {}, acc1 = {};
    #pragma unroll 2
    for (int k = 0; k < D; k += 4) {
      v2f a  = *(const v2f*)(qrow + k);
      v2f b0 = *(const v2f*)(krow0 + k);
      v2f b1 = *(const v2f*)(krow1 + k);
      acc0 = wmma_f32(a, b0, acc0);
      acc1 = wmma_f32(a, b1, acc1);
    }
    #pragma unroll
    for (int v = 0; v < 8; ++v) {
      S[(v + 8 * hf) * N + j0 + m]      = acc0[v] * scale;
      S[(v + 8 * hf) * N + j0 + 16 + m] = acc1[v] * scale;
    }
  }
  __syncthreads();

  // ---- phase 2: row softmax (exp kept unnormalized; keep 1/rowsum) ---
  {
    const int r = tid >> 3, jl = tid & 7;
    float mx = -3.4e38f;
    for (int c2 = jl; c2 < N; c2 += 8) mx = fmaxf(mx, S[r * N + c2]);
    red[r * 8 + jl] = mx;
  }
  __syncthreads();
  if (tid < 16) {
    float mx = red[tid * 8];
    #pragma unroll
    for (int j = 1; j < 8; ++j) mx = fmaxf(mx, red[tid * 8 + j]);
    rmax[tid] = mx;
  }
  __syncthreads();
  {
    const int r = tid >> 3, jl = tid & 7;
    const float mx = rmax[r];
    float sum = 0.f;
    for (int c2 = jl; c2 < N; c2 += 8) {
      const float e = __expf(S[r * N + c2] - mx);
      S[r * N + c2] = e;
      sum += e;
    }
    red[r * 8 + jl] = sum;
  }
  __syncthreads();
  if (tid < 16) {
    float sum = 0.f;
    #pragma unroll
    for (int j = 0; j < 8; ++j) sum += red[tid * 8 + j];
    rinv[tid] = 1.0f / sum;
  }
  __syncthreads();

  // ---- phase 3: Y = P V (paired D tiles), normalize, un-window -------
  const int ytiles = D >> 4;     // min 8 (head 3) = 4 waves * 2
  const int s = 5 - logH;
  for (int yt = wave * 2; yt < ytiles; yt += 8) {
    const int d0 = yt << 4;
    const float* vcol = Vb + d0 + m + (size_t)(2 * hf) * D;
    const float* prow = S + m * N + 2 * hf;
    v8f acc0 = {}, acc1 = {};
    #pragma unroll 2
    for (int kk = 0; kk < N; kk += 4) {
      v2f a = *(const v2f*)(prow + kk);  // ds_load_b64 from LDS, shared
      const float* vp = vcol + (size_t)kk * D;
      v2f b0, b1;
      b0.x = vp[0];
      b0.y = vp[D];
      b1.x = vp[16];
      b1.y = vp[D + 16];
      acc0 = wmma_f32(a, b0, acc0);
      acc1 = wmma_f32(a, b1, acc1);
    }
    #pragma unroll
    for (int dt = 0; dt < 2; ++dt) {
      const v8f& acc = dt ? acc1 : acc0;
      const int dd = d0 + 16 * dt + m;
      const int c = dd >> (2 * logH);
      const int rem2 = dd & ((1 << (2 * logH)) - 1);
      const int hy = rem2 >> logH;
      const int wx = rem2 & ((1 << logH) - 1);
      #pragma unroll
      for (int v = 0; v < 8; ++v) {
        const int r = v + 8 * hf;
        const int wr = q0 + r;
        const int tt = wr >> (2 * s);
        const int rem = wr & ((1 << (2 * s)) - 1);
        const int oh_y = rem >> s;
        const int ow_x = rem & ((1 << s) - 1);
        const int yy = (oh_y << logH) + hy;
        const int xx = (ow_x << logH) + wx;
        const size_t idx =
            (((size_t)(bb * 4 + tt) * 512) + (head * 128 + c)) * 1024 +
            yy * 32 + xx;
        out[idx] = acc[v] * rinv[r];
      }
    }
  }
}

// ---------------------------------------------------------------------------
// Kernel 3: 3x3 SAME conv (implicit GEMM, K ordered (ky,kx,ic)) + bias +
// LeakyReLU(0.2). Each wave computes 32(o) x 32(pix = full row y) as 2x2
// WMMA tiles. grid = (32 rows, 16 o blocks, 16 images), block = 32.
// ---------------------------------------------------------------------------
__global__ __launch_bounds__(32)
void conv3_kernel(const float* __restrict__ in, const float* __restrict__ wo,
                  const float* __restrict__ bo, float* __restrict__ outp) {
  const int lane = threadIdx.x & 31;
  const int m = lane & 15, hf = lane >> 4;
  const int y  = blockIdx.x;          // output row, pix tile covers x=0..31
  const int o0 = blockIdx.y * 32;
  const int n  = blockIdx.z;

  const int x0 = m;                   // lane column, pixel tile 0
  const int x1 = 16 + m;              // lane column, pixel tile 1

  const float* inb   = in + (size_t)n * 512 * 1024;
  const float* arow0 = wo + (size_t)(o0 + m) * 4608;
  const float* arow1 = arow0 + (size_t)16 * 4608;

  v8f acc00 = {}, acc01 = {}, acc10 = {}, acc11 = {};
  #pragma unroll
  for (int ky = 0; ky < 3; ++ky) {
    const int yy = y + ky - 1;
    const bool rok = (yy >= 0) && (yy < 32);
    #pragma unroll
    for (int kx = 0; kx < 3; ++kx) {
      const int seg = ky * 3 + kx;
      const int xs0 = x0 + kx - 1;
      const int xs1 = x1 + kx - 1;
      const bool cok0 = rok && (xs0 >= 0);           // xs0 <= 16 always
      const bool cok1 = rok && (xs1 < 32);           // xs1 >= 15 always
      const int boff0 = cok0 ? (yy * 32 + xs0) : 0;  // clamped valid addr
      const int boff1 = cok1 ? (yy * 32 + xs1) : 0;
      for (int ic = 0; ic < 512; ic += 4) {
        const int ica = ic + 2 * hf;
        v2f a0, a1;
        a0.x = arow0[ica * 9 + seg];
        a0.y = arow0[(ica + 1) * 9 + seg];
        a1.x = arow1[ica * 9 + seg];
        a1.y = arow1[(ica + 1) * 9 + seg];
        const float b0x = inb[(size_t)ica * 1024 + boff0];
        const float b0y = inb[(size_t)(ica + 1) * 1024 + boff0];
        const float b1x = inb[(size_t)ica * 1024 + boff1];
        const float b1y = inb[(size_t)(ica + 1) * 1024 + boff1];
        v2f b0, b1;
        b0.x = cok0 ? b0x : 0.f;
        b0.y = cok0 ? b0y : 0.f;
        b1.x = cok1 ? b1x : 0.f;
        b1.y = cok1 ? b1y : 0.f;
        acc00 = wmma_f32(a0, b0, acc00);
        acc01 = wmma_f32(a0, b1, acc01);
        acc10 = wmma_f32(a1, b0, acc10);
        acc11 = wmma_f32(a1, b1, acc11);
      }
    }
  }
  #pragma unroll
  for (int ot = 0; ot < 2; ++ot) {
    #pragma unroll
    for (int pt = 0; pt < 2; ++pt) {
      const v8f& acc = ot ? (pt ? acc11 : acc10) : (pt ? acc01 : acc00);
      const int xx = 16 * pt + m;
      #pragma unroll
      for (int v = 0; v < 8; ++v) {
        const int o = o0 + 16 * ot + v + 8 * hf;
        float val = acc[v] + bo[o];
        val = (val >= 0.f) ? val : 0.2f * val;
        outp[(((size_t)n * 512 + o) * 32 + y) * 32 + xx] = val;
      }
    }
  }
}

// ---------------------------------------------------------------------------
extern "C" void kernel_launch(void* const* d_in, const int* in_sizes, int n_in,
                              void* d_out, int out_size, void* d_ws,
                              size_t ws_size, hipStream_t stream) {
  const float* x  = (const float*)d_in[0];
  // d_in[1] = mask m: mathematical no-op in the reference, unused.
  const float* wq = (const float*)d_in[2];
  const float* bq = (const float*)d_in[3];
  const float* wk = (const float*)d_in[4];
  const float* bk = (const float*)d_in[5];
  const float* wv = (const float*)d_in[6];
  const float* bv = (const float*)d_in[7];
  const float* wo = (const float*)d_in[8];
  const float* bo = (const float*)d_in[9];

  float* ws = (float*)d_ws;
  float* Qw   = ws + WS_Q;
  float* Kw   = ws + WS_K;
  float* Vw   = ws + WS_V;
  float* Aout = ws + WS_A;

  const dim3 gproj(32, 16, 16);
  qkv_proj_kernel<<<gproj, 32, 0, stream>>>(x, wq, bq, Qw);
  qkv_proj_kernel<<<gproj, 32, 0, stream>>>(x, wk, bk, Kw);
  qkv_proj_kernel<<<gproj, 32, 0, stream>>>(x, wv, bv, Vw);

  for (int head = 0; head < 4; ++head) {
    const int logH = 3 - head;
    const int D = 128 << (2 * logH);
    const int N = 524288 / D;
    const float scale = 1.0f / sqrtf((float)D);
    const size_t lds = (size_t)(16 * N + 128 + 32) * sizeof(float);
    const dim3 gattn(N / 16, 4, 1);
    attn_kernel<<<gattn, 128, lds, stream>>>(
        Qw + (size_t)head * HEAD_STRIDE, Kw + (size_t)head * HEAD_STRIDE,
        Vw + (size_t)head * HEAD_STRIDE, Aout, N, D, head, logH, scale);
  }

  const dim3 gconv(32, 16, 16);
  conv3_kernel<<<gconv, 32, 0, stream>>>(Aout, wo, bo, (float*)d_out);
}